// VoteNetLoss_9740985827846
// MI455X (gfx1250) — compile-verified
//
#include <hip/hip_runtime.h>
#include <hip/hip_bf16.h>
#include <math.h>

typedef __attribute__((ext_vector_type(2))) float v2f;
typedef __attribute__((ext_vector_type(8))) float v8f;

#define B_    64
#define NSEED 1024
#define NPT   20000
#define K_    256
#define K2_   64
#define NH_   12
#define NS_   10
#define NC_   10

__device__ inline float wave_sum(float v) {
#pragma unroll
  for (int off = 16; off > 0; off >>= 1) v += __shfl_xor(v, off, 32);
  return v;
}

__device__ inline float huber1(float e) {
  float a = fabsf(e);
  float q = fminf(a, 1.0f);
  return 0.5f * q * q + (a - q);
}

__device__ inline float ce_loss(const float* __restrict__ logits, int n, int label) {
  float mx = -3.4e38f;
  for (int i = 0; i < n; ++i) mx = fmaxf(mx, logits[i]);
  float s = 0.0f;
  for (int i = 0; i < n; ++i) s += __expf(logits[i] - mx);
  return mx + __logf(s) - logits[label];
}

// ---------------------------------------------------------------------------
// accumulator slots (floats at start of ws):
// 0 vote_num  1 vote_den  2 obj_num  3 obj_den  4 objf_sum 5 d1c_num
// 6 blm_sum   7 d2c_num   8 hcls     9 hreg    10 scls    11 sreg   12 sem
// ---------------------------------------------------------------------------

__global__ void zero_kernel(float* acc, int n) {
  int t = blockIdx.x * blockDim.x + threadIdx.x;
  if (t < n) acc[t] = 0.0f;
}

// --- vote loss: gather + min-of-3 L1 distances, masked mean -----------------
__global__ __launch_bounds__(256) void vote_loss_kernel(
    const float* __restrict__ seed_xyz, const float* __restrict__ vote_xyz,
    const float* __restrict__ vote_label, const int* __restrict__ seed_inds,
    const int* __restrict__ vote_label_mask, float* __restrict__ acc) {
  int t = blockIdx.x * blockDim.x + threadIdx.x;
  float num = 0.0f, den = 0.0f;
  if (t < B_ * NSEED) {
    int b = t / NSEED;
    int idx = seed_inds[t];
    float m = (float)vote_label_mask[(size_t)b * NPT + idx];
    const float* gl = vote_label + ((size_t)b * NPT + idx) * 9;
    float sx = seed_xyz[t * 3 + 0], sy = seed_xyz[t * 3 + 1], sz = seed_xyz[t * 3 + 2];
    float vx = vote_xyz[t * 3 + 0], vy = vote_xyz[t * 3 + 1], vz = vote_xyz[t * 3 + 2];
    float best = 3.4e38f;
#pragma unroll
    for (int j = 0; j < 3; ++j) {
      float gx = gl[j * 3 + 0] + sx;
      float gy = gl[j * 3 + 1] + sy;
      float gz = gl[j * 3 + 2] + sz;
      float d = fabsf(vx - gx) + fabsf(vy - gy) + fabsf(vz - gz);
      best = fminf(best, d);
    }
    num = best * m;
    den = m;
  }
  num = wave_sum(num);
  den = wave_sum(den);
  if ((threadIdx.x & 31) == 0) {
    atomicAdd(&acc[0], num);
    atomicAdd(&acc[1], den);
  }
}

// --- WMMA nn_distance row-min: P(B,NP,3) vs Q(B,NQ,3) -----------------------
// sqdist(m,n) = |p_m|^2 + |q_n|^2 - 2 p_m.q_n ; cross-term via
// V_WMMA_F32_16X16X4_F32 with K padded 3 -> 4. One wave = one 16-row tile.
// NP/NQ compile-time so the NQ/16 column-tile loop fully unrolls: all four
// B-fragments + column norms load up front (one wait), then 4 back-to-back
// WMMAs, then fused min/argmin in registers (no dist matrix in memory).
template <int NP, int NQ, bool WRITE_IDX>
__global__ __launch_bounds__(32) void wmma_nn_rowmin_kernel(
    const float* __restrict__ P, const float* __restrict__ Q,
    float* __restrict__ outMin, int* __restrict__ outIdx) {
  constexpr int NT = NQ / 16;
  const int tilesPerB = NP / 16;
  int b  = blockIdx.x / tilesPerB;
  int m0 = (blockIdx.x % tilesPerB) * 16;
  int lane = threadIdx.x;
  int half = lane >> 4;   // 0: K={0,1} / M=v ; 1: K={2,3} / M=v+8
  int lm   = lane & 15;

  const float* Pb = P + (size_t)b * NP * 3;
  const float* Qb = Q + (size_t)b * NQ * 3;

  // A fragment: row m0+lm, (x,y) in low-half lanes, (z,0) in high-half lanes
  int rowA = m0 + lm;
  float ax = Pb[rowA * 3 + 0], ay = Pb[rowA * 3 + 1], az = Pb[rowA * 3 + 2];
  v2f a;
  a.x = half ? az : ax;
  a.y = half ? 0.0f : ay;

  // All B fragments + column norms, loads issued up front
  v2f bb[NT];
  float qn[NT];
#pragma unroll
  for (int nt = 0; nt < NT; ++nt) {
    int col = nt * 16 + lm;
    float qx = Qb[col * 3 + 0], qy = Qb[col * 3 + 1], qz = Qb[col * 3 + 2];
    bb[nt].x = half ? qz : qx;    // B rows K=0 (x) / K=2 (z) in VGPR0 halves,
    bb[nt].y = half ? 0.0f : qy;  //        K=1 (y) / K=3 (0) in VGPR1 halves
    qn[nt] = qx * qx + qy * qy + qz * qz;
  }

  // row norms for the 8 C rows this lane covers (M = m0 + 8*half + v)
  float rn[8];
#pragma unroll
  for (int v = 0; v < 8; ++v) {
    int r = m0 + half * 8 + v;
    float x = Pb[r * 3 + 0], y = Pb[r * 3 + 1], z = Pb[r * 3 + 2];
    rn[v] = x * x + y * y + z * z;
  }

  float minv[8];
  int mini[8];
#pragma unroll
  for (int v = 0; v < 8; ++v) { minv[v] = 3.4e38f; mini[v] = 0; }

#pragma unroll
  for (int nt = 0; nt < NT; ++nt) {
    v8f c = {0.f, 0.f, 0.f, 0.f, 0.f, 0.f, 0.f, 0.f};
    c = __builtin_amdgcn_wmma_f32_16x16x4_f32(false, a, false, bb[nt],
                                              (short)0, c, false, false);
#pragma unroll
    for (int v = 0; v < 8; ++v) {
      float d = rn[v] + qn[nt] - 2.0f * c[v];
      int col = nt * 16 + lm;
      if (d < minv[v]) { minv[v] = d; mini[v] = col; }
    }
  }

  // reduce min/argmin across the 16 lanes of each half (xor stays in-half)
#pragma unroll
  for (int off = 8; off >= 1; off >>= 1) {
#pragma unroll
    for (int v = 0; v < 8; ++v) {
      float ov = __shfl_xor(minv[v], off, 32);
      int   oi = __shfl_xor(mini[v], off, 32);
      if (ov < minv[v] || (ov == minv[v] && oi < mini[v])) {
        minv[v] = ov; mini[v] = oi;
      }
    }
  }
  if (lm == 0) {
#pragma unroll
    for (int v = 0; v < 8; ++v) {
      int r = m0 + half * 8 + v;
      outMin[(size_t)b * NP + r] = minv[v];
      if (WRITE_IDX) outIdx[(size_t)b * NP + r] = mini[v];
    }
  }
}

// --- d2c: per-GT-center min over K proposals, masked by box_label_mask ------
__global__ __launch_bounds__(256) void d2c_kernel(
    const float* __restrict__ center, const float* __restrict__ gt,
    const int* __restrict__ blm, float* __restrict__ acc) {
  int t = blockIdx.x * blockDim.x + threadIdx.x;
  float num = 0.0f, den = 0.0f;
  if (t < B_ * K2_) {
    int b = t / K2_;
    float gx = gt[t * 3 + 0], gy = gt[t * 3 + 1], gz = gt[t * 3 + 2];
    const float* cb = center + (size_t)b * K_ * 3;
    float best = 3.4e38f;
    for (int i = 0; i < K_; ++i) {
      float dx = cb[i * 3 + 0] - gx;
      float dy = cb[i * 3 + 1] - gy;
      float dz = cb[i * 3 + 2] - gz;
      best = fminf(best, dx * dx + dy * dy + dz * dz);
    }
    float m = (float)blm[t];
    num = best * m;
    den = m;
  }
  num = wave_sum(num);
  den = wave_sum(den);
  if ((threadIdx.x & 31) == 0) {
    atomicAdd(&acc[7], num);
    atomicAdd(&acc[6], den);
  }
}

// --- objectness + all assign-gathered losses --------------------------------
__global__ __launch_bounds__(256) void main_losses_kernel(
    const float* __restrict__ dist1, const int* __restrict__ ind1,
    const float* __restrict__ d1c, const float* __restrict__ obj_scores,
    const float* __restrict__ heading_scores, const float* __restrict__ hres_norm,
    const float* __restrict__ hres_label, const float* __restrict__ size_scores,
    const float* __restrict__ sres_norm, const float* __restrict__ sres_label,
    const float* __restrict__ mean_size, const float* __restrict__ sem_scores,
    const int* __restrict__ hcls_label, const int* __restrict__ scls_label,
    const int* __restrict__ sem_label, float* __restrict__ acc) {
  int t = blockIdx.x * blockDim.x + threadIdx.x;
  float s_objnum = 0, s_objden = 0, s_objf = 0, s_d1c = 0;
  float s_hc = 0, s_hr = 0, s_sc = 0, s_sr = 0, s_sem = 0;
  if (t < B_ * K_) {
    int b = t / K_;
    int a = ind1[t];

    float euclid = sqrtf(dist1[t] + 1e-6f);
    int obj = (euclid < 0.1f) ? 1 : 0;
    float maskv = ((euclid < 0.1f) || (euclid > 0.4f)) ? 1.0f : 0.0f;
    float objf = (float)obj;

    // 2-class weighted CE (OBJ_W = {0.1, 0.9})
    {
      float l0 = obj_scores[t * 2 + 0], l1 = obj_scores[t * 2 + 1];
      float mx = fmaxf(l0, l1);
      float lse = mx + __logf(__expf(l0 - mx) + __expf(l1 - mx));
      float ce = lse - (obj ? l1 : l0);
      float w = obj ? 0.9f : 0.1f;
      s_objnum = ce * w * maskv;
      s_objden = maskv;
    }
    s_objf = objf;
    s_d1c  = d1c[t] * objf;

    // heading classification + residual
    int hcl = hcls_label[b * K2_ + a];
    s_hc = ce_loss(heading_scores + (size_t)t * NH_, NH_, hcl) * objf;
    {
      float hrl = hres_label[b * K2_ + a] * ((float)NH_ / (float)M_PI);
      float pred = hres_norm[(size_t)t * NH_ + hcl];
      s_hr = huber1(pred - hrl) * objf;
    }

    // size classification + residual
    int scl = scls_label[b * K2_ + a];
    s_sc = ce_loss(size_scores + (size_t)t * NS_, NS_, scl) * objf;
    {
      const float* srl = sres_label + ((size_t)b * K2_ + a) * 3;
      const float* pr  = sres_norm + ((size_t)t * NS_ + scl) * 3;
      const float* ms  = mean_size + scl * 3;
      float s = 0.0f;
#pragma unroll
      for (int c = 0; c < 3; ++c) s += huber1(pr[c] - srl[c] / ms[c]);
      s_sr = (s * (1.0f / 3.0f)) * objf;
    }

    // semantic class
    int sem = sem_label[b * K2_ + a];
    s_sem = ce_loss(sem_scores + (size_t)t * NC_, NC_, sem) * objf;
  }

  s_objnum = wave_sum(s_objnum); s_objden = wave_sum(s_objden);
  s_objf = wave_sum(s_objf);     s_d1c = wave_sum(s_d1c);
  s_hc = wave_sum(s_hc);         s_hr = wave_sum(s_hr);
  s_sc = wave_sum(s_sc);         s_sr = wave_sum(s_sr);
  s_sem = wave_sum(s_sem);
  if ((threadIdx.x & 31) == 0) {
    atomicAdd(&acc[2], s_objnum);
    atomicAdd(&acc[3], s_objden);
    atomicAdd(&acc[4], s_objf);
    atomicAdd(&acc[5], s_d1c);
    atomicAdd(&acc[8], s_hc);
    atomicAdd(&acc[9], s_hr);
    atomicAdd(&acc[10], s_sc);
    atomicAdd(&acc[11], s_sr);
    atomicAdd(&acc[12], s_sem);
  }
}

__global__ void finalize_kernel(const float* __restrict__ acc, float* __restrict__ out) {
  if (threadIdx.x == 0 && blockIdx.x == 0) {
    const float eps = 1e-6f;
    float vote = acc[0] / (acc[1] + eps);
    float obj  = acc[2] / (acc[3] + eps);
    float objf = acc[4];
    float center = acc[5] / (objf + eps) + acc[7] / (acc[6] + eps);
    float hc = acc[8] / (objf + eps);
    float hr = acc[9] / (objf + eps);
    float sc = acc[10] / (objf + eps);
    float sr = acc[11] / (objf + eps);
    float sem = acc[12] / (objf + eps);
    float box = center + 0.1f * hc + hr + 0.1f * sc + sr;
    out[0] = (vote + 0.5f * obj + box + 0.1f * sem) * 10.0f;
  }
}

extern "C" void kernel_launch(void* const* d_in, const int* in_sizes, int n_in,
                              void* d_out, int out_size, void* d_ws, size_t ws_size,
                              hipStream_t stream) {
  const float* seed_xyz   = (const float*)d_in[0];
  const float* vote_xyz   = (const float*)d_in[1];
  const float* vote_label = (const float*)d_in[2];
  const float* agg_xyz    = (const float*)d_in[3];
  const float* center_lbl = (const float*)d_in[4];   // gt_center
  const float* obj_scores = (const float*)d_in[5];
  const float* center     = (const float*)d_in[6];
  const float* hres_label = (const float*)d_in[7];
  const float* hscores    = (const float*)d_in[8];
  const float* hres_norm  = (const float*)d_in[9];
  const float* sres_label = (const float*)d_in[10];
  const float* sscores    = (const float*)d_in[11];
  const float* sres_norm  = (const float*)d_in[12];
  const float* sem_scores = (const float*)d_in[13];
  const float* mean_size  = (const float*)d_in[14];
  const int* seed_inds    = (const int*)d_in[15];
  const int* vlm          = (const int*)d_in[16];
  const int* blm          = (const int*)d_in[17];
  const int* hcls_label   = (const int*)d_in[18];
  const int* scls_label   = (const int*)d_in[19];
  const int* sem_label    = (const int*)d_in[20];

  // workspace layout
  float* acc   = (float*)d_ws;              // 16 accumulator floats
  float* dist1 = acc + 16;                  // B*K
  int*   ind1  = (int*)(dist1 + B_ * K_);   // B*K
  float* d1c   = (float*)(ind1 + B_ * K_);  // B*K

  zero_kernel<<<1, 32, 0, stream>>>(acc, 16);

  vote_loss_kernel<<<(B_ * NSEED + 255) / 256, 256, 0, stream>>>(
      seed_xyz, vote_xyz, vote_label, seed_inds, vlm, acc);

  // dist1/ind1: aggregated_vote_xyz vs gt_center (WMMA cross-term)
  wmma_nn_rowmin_kernel<K_, K2_, true><<<B_ * (K_ / 16), 32, 0, stream>>>(
      agg_xyz, center_lbl, dist1, ind1);

  // d1c: center vs gt_center (row mins only)
  wmma_nn_rowmin_kernel<K_, K2_, false><<<B_ * (K_ / 16), 32, 0, stream>>>(
      center, center_lbl, d1c, nullptr);

  d2c_kernel<<<(B_ * K2_ + 255) / 256, 256, 0, stream>>>(center, center_lbl, blm, acc);

  main_losses_kernel<<<(B_ * K_ + 255) / 256, 256, 0, stream>>>(
      dist1, ind1, d1c, obj_scores, hscores, hres_norm, hres_label,
      sscores, sres_norm, sres_label, mean_size, sem_scores,
      hcls_label, scls_label, sem_label, acc);

  finalize_kernel<<<1, 1, 0, stream>>>(acc, (float*)d_out);
}